// Cubify_22986664968359
// MI455X (gfx1250) — compile-verified
//
#include <hip/hip_runtime.h>

// ---------------- problem constants ----------------
#define NB   8
#define DD   48
#define HH   48
#define WW   48
#define HW   (HH * WW)            // 2304
#define DHW  (DD * HW)            // 110592
#define W1   (WW + 1)             // 49
#define H1   (HH + 1)             // 49
#define NV   ((DD + 1) * (HH + 1) * (WW + 1))   // 117649

// output layout (floats), in reference return order
#define SZ_FMASK ((size_t)NB * 6 * DHW)         // 5,308,416
#define OFF_FACES (SZ_FMASK)
#define SZ_FACES ((size_t)6 * DHW * 6)          // 3,981,312
#define OFF_VUSED (OFF_FACES + SZ_FACES)        // 9,289,728
#define SZ_VUSED ((size_t)NB * NV)              // 941,192
#define OFF_VPOS (OFF_VUSED + SZ_VUSED)         // 10,230,920

#define THRESH 0.5f

// _CORNERS from the reference: face f -> 4 (dz,dy,dx) corner offsets.
__device__ __constant__ int c_corners[6][4][3] = {
    {{0,0,0},{0,0,1},{0,1,0},{0,1,1}},   // f0: -z
    {{1,0,0},{1,0,1},{1,1,0},{1,1,1}},   // f1: +z
    {{1,0,0},{1,0,1},{0,0,0},{0,0,1}},   // f2: -y
    {{0,1,0},{0,1,1},{1,1,0},{1,1,1}},   // f3: +y
    {{1,0,0},{0,0,0},{1,1,0},{0,1,0}},   // f4: -x
    {{0,0,1},{1,0,1},{0,1,1},{1,1,1}}    // f5: +x
};

// ---------------- gfx1250 async-to-LDS plumbing (guarded) ----------------
#if __has_builtin(__builtin_amdgcn_global_load_async_to_lds_b128)
#define CUBIFY_ASYNC 1
#else
#define CUBIFY_ASYNC 0
#endif

// Per hipcc diagnostic, the builtin's params are int4-vector pointers:
//   arg0: v4i in addrspace(1) (global src), arg1: v4i in addrspace(3) (LDS dst)
typedef int v4i_t __attribute__((ext_vector_type(4)));
typedef __attribute__((address_space(1))) v4i_t* gptr128_t;
typedef __attribute__((address_space(3))) v4i_t* lptr128_t;

__device__ __forceinline__ void cubify_wait_async0() {
#if __has_builtin(__builtin_amdgcn_s_wait_asynccnt)
    __builtin_amdgcn_s_wait_asynccnt(0);
#else
    asm volatile("s_wait_asynccnt 0" ::: "memory");
#endif
}

// ---------------- kernel 1: data-independent outputs + zero init ----------------
// faces (as float), vert_pos, and zero vert_used (scatter target of kernel 2).
__global__ void cubify_static(float* __restrict__ out) {
    const int tid = blockIdx.x * blockDim.x + threadIdx.x;
    const int nthreads = gridDim.x * blockDim.x;

    // faces: 6*DHW cells, each emits 6 floats: ids[0],ids[1],ids[2], ids[1],ids[2],ids[3]
    for (int i = tid; i < 6 * DHW; i += nthreads) {
        const int f  = i / DHW;
        const int r  = i % DHW;
        const int z  = r / HW;
        const int r2 = r % HW;
        const int y  = r2 / WW;
        const int x  = r2 % WW;
        int id[4];
#pragma unroll
        for (int c = 0; c < 4; ++c) {
            const int vz = z + c_corners[f][c][0];
            const int vy = y + c_corners[f][c][1];
            const int vx = x + c_corners[f][c][2];
            id[c] = (vz * H1 + vy) * W1 + vx;
        }
        float* o = out + OFF_FACES + (size_t)i * 6;
        o[0] = (float)id[0]; o[1] = (float)id[1]; o[2] = (float)id[2];
        o[3] = (float)id[1]; o[4] = (float)id[2]; o[5] = (float)id[3];
    }

    // vert_pos: (vz,vy,vx) - 0.5
    for (int v = tid; v < NV; v += nthreads) {
        const int vz  = v / (H1 * W1);
        const int rem = v % (H1 * W1);
        const int vy  = rem / W1;
        const int vx  = rem % W1;
        float* o = out + OFF_VPOS + (size_t)v * 3;
        o[0] = (float)vz - 0.5f;
        o[1] = (float)vy - 0.5f;
        o[2] = (float)vx - 0.5f;
    }

    // zero vert_used (re-done every call; harness does not re-poison)
    for (size_t i = tid; i < SZ_VUSED; i += (size_t)nthreads) {
        out[OFF_VUSED + i] = 0.0f;
    }
}

// ---------------- kernel 2: face_mask + vert_used scatter ----------------
#define TILE_Y   8
#define NTHREADS 384            // 8 rows * 48 x = 12 wave32s

__global__ __launch_bounds__(NTHREADS)
void cubify_faces(const float* __restrict__ vox, float* __restrict__ out) {
    // (z-1,z,z+1) x (ybase-1 .. ybase+TILE_Y) x full x row, fp32
    __shared__ float tile[3][TILE_Y + 2][WW];

    const int b     = blockIdx.x;
    const int YT    = HH / TILE_Y;                 // 6
    const int n     = b / (DD * YT);
    const int r     = b % (DD * YT);
    const int z     = r / YT;
    const int ybase = (r % YT) * TILE_Y;
    const float* in_n = vox + (size_t)n * DHW;

    // ---- stage halo tile into LDS in 16B chunks ----
    const int CH_PER_ROW = WW / 4;                          // 12 float4 per row
    const int NCH = 3 * (TILE_Y + 2) * CH_PER_ROW;          // 360 chunks
    for (int j = threadIdx.x; j < NCH; j += NTHREADS) {
        const int zl = j / ((TILE_Y + 2) * CH_PER_ROW);
        const int rr = j % ((TILE_Y + 2) * CH_PER_ROW);
        const int yl = rr / CH_PER_ROW;
        const int xc = (rr % CH_PER_ROW) * 4;
        const int gz = z - 1 + zl;
        const int gy = ybase - 1 + yl;
        const bool inr = (gz >= 0) & (gz < DD) & (gy >= 0) & (gy < HH);
        float* dst = &tile[zl][yl][xc];
        if (inr) {
            float* src = const_cast<float*>(in_n + (size_t)(gz * HH + gy) * WW + xc);
#if CUBIFY_ASYNC
            __builtin_amdgcn_global_load_async_to_lds_b128(
                (gptr128_t)(v4i_t*)src, (lptr128_t)(v4i_t*)dst, 0, 0);
#else
            *(float4*)dst = *(const float4*)src;
#endif
        } else {
            *(float4*)dst = make_float4(0.f, 0.f, 0.f, 0.f);  // padded boundary = unoccupied
        }
    }
#if CUBIFY_ASYNC
    cubify_wait_async0();
#endif
    __syncthreads();

    // ---- compute 6 face masks for this thread's voxel ----
    const int x  = threadIdx.x % WW;
    const int y0 = threadIdx.x / WW;      // 0..7
    const int y  = ybase + y0;

    const bool occ = tile[1][y0 + 1][x] > THRESH;
    const bool nb0 = tile[0][y0 + 1][x] > THRESH;                       // z-1
    const bool nb1 = tile[2][y0 + 1][x] > THRESH;                       // z+1
    const bool nb2 = tile[1][y0    ][x] > THRESH;                       // y-1
    const bool nb3 = tile[1][y0 + 2][x] > THRESH;                       // y+1
    const bool nb4 = (x > 0)      ? (tile[1][y0 + 1][x - 1] > THRESH) : false;  // x-1
    const bool nb5 = (x < WW - 1) ? (tile[1][y0 + 1][x + 1] > THRESH) : false;  // x+1

    bool fm[6];
    fm[0] = occ && !nb0;  fm[1] = occ && !nb1;
    fm[2] = occ && !nb2;  fm[3] = occ && !nb3;
    fm[4] = occ && !nb4;  fm[5] = occ && !nb5;

    const size_t vox_off = (size_t)z * HW + (size_t)y * WW + x;
#pragma unroll
    for (int f = 0; f < 6; ++f) {
        out[(size_t)(n * 6 + f) * DHW + vox_off] = fm[f] ? 1.0f : 0.0f;
    }

    // scatter vert_used: mark 4 corner ids of every exposed face.
    // Races write the identical constant 1.0f -> deterministic.
    float* vu = out + OFF_VUSED + (size_t)n * NV;
#pragma unroll
    for (int f = 0; f < 6; ++f) {
        if (fm[f]) {
#pragma unroll
            for (int c = 0; c < 4; ++c) {
                const int id = ((z + c_corners[f][c][0]) * H1 +
                                (y + c_corners[f][c][1])) * W1 +
                                (x + c_corners[f][c][2]);
                vu[id] = 1.0f;
            }
        }
    }
}

// ---------------- launch ----------------
extern "C" void kernel_launch(void* const* d_in, const int* in_sizes, int n_in,
                              void* d_out, int out_size, void* d_ws, size_t ws_size,
                              hipStream_t stream) {
    (void)in_sizes; (void)n_in; (void)out_size; (void)d_ws; (void)ws_size;
    const float* vox = (const float*)d_in[0];
    float* out = (float*)d_out;

    // kernel 1: faces + vert_pos + zero vert_used (6*DHW = 663552 = 2592*256)
    cubify_static<<<2592, 256, 0, stream>>>(out);

    // kernel 2: one block per (n, z, 8-row y tile): 8*48*6 = 2304 blocks
    cubify_faces<<<NB * DD * (HH / TILE_Y), NTHREADS, 0, stream>>>(vox, out);
}